// CoCovTranspose_65360812310719
// MI455X (gfx1250) — compile-verified
//
#include <hip/hip_runtime.h>
#include <hip/hip_bf16.h>
#include <stdint.h>

typedef float v2f  __attribute__((ext_vector_type(2)));
typedef float v8f  __attribute__((ext_vector_type(8)));
typedef unsigned int u32x4 __attribute__((ext_vector_type(4)));
typedef int   i32x4 __attribute__((ext_vector_type(4)));
typedef int   i32x8 __attribute__((ext_vector_type(8)));

#define HALO 96     // staged input span per tile (3 x 32)
#define NK   81     // 9 stacked kernels * 9 taps

// One workgroup = one 64x64 output tile of one batch image.
// 256 threads = 8 wave32. LDS: input halo 288KB + weights 20.25KB + bias.
__global__ __launch_bounds__(256, 1)
void cocov_transpose_kernel(const float* __restrict__ inp,    // (8,8,512,512)
                            const float* __restrict__ wts,    // (9,8,8,3,3)
                            const float* __restrict__ bias,   // (9,8)
                            float* __restrict__ out)          // (8,8,1024,1024)
{
    __shared__ __align__(16) float sIn[8 * HALO * HALO]; // [ci][y][x]  294912 B
    __shared__ __align__(16) float sW[NK * 64];          // [(k*3+kh)*3+kw][ci][co]
    __shared__ float sB[8];                              // summed bias per co

    const int j    = blockIdx.x;        // tile col  [0,16)
    const int i    = blockIdx.y;        // tile row  [0,16)
    const int b    = blockIdx.z;        // batch     [0,8)
    const int tid  = threadIdx.x;
    const int lane = tid & 31;
    const int wv   = tid >> 5;
    const int n    = lane & 15;         // WMMA column lane (N), also co index for A
    const int csel = lane >> 4;         // K half select: lanes 0-15 -> K{0,1}, 16-31 -> K{2,3}

    // ---- stage weights reordered to [w33][ci][co] and summed bias ----
    for (int idx = tid; idx < NK * 64; idx += 256) {
        const int co  = idx & 7;
        const int ci  = (idx >> 3) & 7;
        const int w33 = idx >> 6;                 // (k*3+kh)*3+kw
        const int k   = w33 / 9;
        const int khw = w33 % 9;                  // kh*3+kw, contiguous in source
        sW[idx] = wts[((k * 8 + ci) * 8 + co) * 9 + khw];
    }
    if (tid < 8) {
        float s = 0.f;
        for (int k = 0; k < 9; ++k) s += bias[k * 8 + tid];
        sB[tid] = s;
    }

    // ---- stage 96x96x8ci input halo into LDS ----
    const int gy0 = i * 32 - 32;
    const int gx0 = j * 32 - 32;
    const bool interior = (i >= 1) && (i <= 14) && (j >= 1) && (j <= 14);
    bool staged = false;

#if __has_builtin(__builtin_amdgcn_tensor_load_to_lds)
    if (interior) {
        if (wv == 0) {
            // Tensor Data Mover: 3D tile x=96, y=96, z=8(ci) dwords, packed into LDS.
            const uint32_t lds0 = (uint32_t)(uintptr_t)&sIn[0];       // low 32 bits = LDS offset
            const uint64_t ga = (uint64_t)(uintptr_t)inp
                + 4ull * ((uint64_t)(b * 8) * 512u * 512u + (uint64_t)gy0 * 512u + (uint64_t)gx0);
            u32x4 g0 = { 1u,                                           // count=1 (valid D#)
                         lds0,                                         // lds_addr
                         (uint32_t)ga,                                 // global_addr lo
                         (uint32_t)((ga >> 32) & 0x1ffffffu) | (2u << 30) }; // addr hi | type=2
            i32x8 g1 = { (int)(2u << 16),            // data_size = 4B, no multicast/pad
                         (int)(512u << 16),          // tensor_dim0[15:0]=512 (bits 79:48)
                         (int)(512u << 16),          // tensor_dim1[15:0]=512 (bits 111:80)
                         (int)(96u << 16),           // tile_dim0 = 96
                         (int)((8u << 16) | 96u),    // tile_dim2=8 | tile_dim1=96
                         512,                        // tensor_dim0_stride = 512 dwords
                         0,                          // stride0 hi / stride1 lo16 = 0
                         4 };                        // tensor_dim1_stride = 512*512 (>>16)
            i32x4 g2 = { 8, 1, 0, 0 };               // tensor_dim2=8, tensor_dim3=1
            i32x4 g3 = { 0, 0, 0, 0 };
            i32x8 g4 = { 0, 0, 0, 0, 0, 0, 0, 0 };   // extra operand (clang-23 6-arg form)
            __builtin_amdgcn_tensor_load_to_lds(g0, g1, g2, g3, g4, 0);
#if __has_builtin(__builtin_amdgcn_s_wait_tensorcnt)
            __builtin_amdgcn_s_wait_tensorcnt(0);
#else
            asm volatile("s_wait_tensorcnt 0" ::: "memory");
#endif
        }
        staged = true;
    }
#endif
    if (!staged) {
        // boundary tiles: manual float4 staging with zero fill outside the image
        for (int f = tid; f < 8 * HALO * (HALO / 4); f += 256) {
            const int x4 = f % 24;
            const int y  = (f / 24) % HALO;
            const int ci = f / (24 * HALO);
            const int gx = gx0 + x4 * 4;
            const int gy = gy0 + y;
            float4 v = make_float4(0.f, 0.f, 0.f, 0.f);
            if ((unsigned)gy < 512u && (unsigned)gx < 509u)
                v = *(const float4*)&inp[(((size_t)b * 8 + ci) * 512 + gy) * 512 + gx];
            *(float4*)&sIn[(ci * HALO + y) * HALO + x4 * 4] = v;
        }
    }
    __syncthreads();

    // Force the wave index into an SGPR so every chunk/tap condition derived
    // from it is provably wave-uniform -> scalar branches, EXEC stays all-ones
    // around the WMMAs (ISA requirement), no saveexec divergence handling.
    const int wvs = __builtin_amdgcn_readfirstlane(wv);

    // Per-lane clamped indices for operand fetch (loads always in bounds;
    // zeroing done with cndmask selects after the load).
    const int nco = n & 7;              // A row within real co range

    // D = A(16x4: M=co, K=ci slice) x B(4x16: K=ci slice, N=pixels) + C
    // A lane layout: v[0]=A[M=n][K=2*csel], v[1]=A[M=n][K=2*csel+1]; B mirrored.
    auto mm = [&](v8f c, int w33, int yl, int p0, int xb) -> v8f {
        const int  p  = p0 + n;                       // per-lane input x within tile
        const bool pv = (p < 32);                     // transposed-conv padding crop
        const int  xi = xb + (pv ? p : 31);           // clamped, always in bounds
        const int  bbase = yl * HALO + xi;            // + ci*HALO*HALO below
#pragma unroll
        for (int ci4 = 0; ci4 < 8; ci4 += 4) {
            const int ci = ci4 + 2 * csel;
            const int wb = (w33 * 8 + ci) * 8;
            const float w0 = sW[wb + nco];            // unconditional LDS loads
            const float w1 = sW[wb + 8 + nco];
            const float t0 = sIn[ci * (HALO * HALO) + bbase];
            const float t1 = sIn[(ci + 1) * (HALO * HALO) + bbase];
            const float a0 = (n < 8) ? w0 : 0.f;      // cndmask, not branches
            const float a1 = (n < 8) ? w1 : 0.f;
            const float b0 = pv ? t0 : 0.f;
            const float b1 = pv ? t1 : 0.f;
            const v2f A = { a0, a1 };
            const v2f B = { b0, b1 };
            c = __builtin_amdgcn_wmma_f32_16x16x4_f32(false, A, false, B,
                                                      (short)0, c, false, false);
        }
        return c;
    };

    // 128 chunk-pairs per tile: (yo in [0,64)) x (x half). Each wave takes 16.
    for (int t = 0; t < 16; ++t) {
        const int pid = wvs * 16 + t;       // SGPR
        const int yo  = pid >> 1;           // SGPR
        const int x0  = (pid & 1) << 5;     // SGPR: 0 or 32
        const int ph  = x0 >> 1;

        v8f aE, aO;                          // even-x / odd-x pixel accumulators
#pragma unroll
        for (int r = 0; r < 8; ++r) {
            const float bz = (csel == 0) ? sB[r] : 0.f;   // bias only on real co rows
            aE[r] = bz;
            aO[r] = bz;
        }

        const int yopar = yo & 1;            // SGPR
        for (int k = 0; k < 9; ++k) {
            const int kr = k / 3, kc = k % 3;
            const int xb = kc * 32;
#pragma unroll
            for (int kh = 0; kh < 3; ++kh) {
                if ((kh & 1) == yopar) continue;          // uniform (scalar) skip
                const int q = (yo + 1 - kh) >> 1;
                if ((unsigned)q >= 32u) continue;         // uniform: only yo==63,kh==0
                const int yl = kr * 32 + q;
                const int w3 = (k * 3 + kh) * 3;
                aE = mm(aE, w3 + 1, yl, ph,     xb);      // even x: kw=1
                aO = mm(aO, w3 + 0, yl, ph + 1, xb);      // odd  x: kw=0
                aO = mm(aO, w3 + 2, yl, ph,     xb);      // odd  x: kw=2
            }
        }

        if (csel == 0) {                                   // lanes 0-15 hold co=r rows
            const int gy = i * 64 + yo;
            const int gx = j * 64 + x0 + 2 * n;
#pragma unroll
            for (int r = 0; r < 8; ++r) {
                float2 v = make_float2(aE[r], aO[r]);      // adjacent even/odd pixels
                *(float2*)&out[(((size_t)b * 8 + r) * 1024 + gy) * 1024 + gx] = v;
            }
        }
    }
}

extern "C" void kernel_launch(void* const* d_in, const int* in_sizes, int n_in,
                              void* d_out, int out_size, void* d_ws, size_t ws_size,
                              hipStream_t stream) {
    (void)in_sizes; (void)n_in; (void)out_size; (void)d_ws; (void)ws_size;
    const float* inp  = (const float*)d_in[0];
    const float* wts  = (const float*)d_in[1];
    const float* bias = (const float*)d_in[2];
    float* out = (float*)d_out;
    dim3 grid(16, 16, 8), block(256);
    hipLaunchKernelGGL(cocov_transpose_kernel, grid, block, 0, stream,
                       inp, wts, bias, out);
}